// IterNormRotation_3899830304788
// MI455X (gfx1250) — compile-verified
//
#include <hip/hip_runtime.h>
#include <hip/hip_bf16.h>

#define EPS 1e-5f

typedef float v2f __attribute__((ext_vector_type(2)));
typedef float v8f __attribute__((ext_vector_type(8)));

// ---- workspace layout (float offsets) ----
#define WS_MEAN_ACC 0        // 128
#define WS_GRAM     128      // 16384
#define WS_MEAN     16512    // 128
#define WS_SN       16640    // 16384
#define WS_P        33024    // 16384
#define WS_T1       49408    // 16384
#define WS_T2       65792    // 16384
#define WS_WF       82176    // 16384 (W in B-fragment order, see ns_kernel)
#define WS_SCAL     98560    // 1 (sqrt(rTr))

__device__ inline v8f v8zero() {
  v8f z;
#pragma unroll
  for (int r = 0; r < 8; ++r) z[r] = 0.f;
  return z;
}

// D = A(16x4) * B(4x16) + C, fp32 WMMA
__device__ inline v8f wmma4(v2f a, v2f b, v8f c) {
  return __builtin_amdgcn_wmma_f32_16x16x4_f32(
      false, a, false, b, (short)0, c, false, false);
}

__device__ inline void gsync() { __threadfence(); __syncthreads(); }

// =====================================================================
// Kernel 1: Gram = X^T X  (accumulated in registers, atomic-added at end)
//           + column sums for the mean.
// block = 256 threads (8 waves). Each wave owns tile-row w (16 rows of Gram),
// all 8 tile-cols -> 64 fp32 accum VGPRs persistent across the whole pass.
// =====================================================================
__global__ __launch_bounds__(256) void gram_mean_kernel(
    const float* __restrict__ X, float* __restrict__ ws, int nChunks)
{
  float* gram     = ws + WS_GRAM;
  float* mean_acc = ws + WS_MEAN_ACC;

  __shared__ float sX[32 * 136];   // 32 rows x 128 cols, pad 136 (bank-safe)
  __shared__ float sm[128];

  const int t    = threadIdx.x;
  const int w    = t >> 5;          // wave 0..7
  const int lane = t & 31;
  const int l15  = lane & 15;
  const int h    = lane >> 4;       // K half select
  const int col4 = t & 31;          // this thread's fixed float4 column

  if (t < 128) sm[t] = 0.f;
  __syncthreads();

  v8f acc[8];
#pragma unroll
  for (int j = 0; j < 8; ++j) acc[j] = v8zero();
  float4 csum = make_float4(0.f, 0.f, 0.f, 0.f);

  const float4* X4 = (const float4*)X;

  for (int c = blockIdx.x; c < nChunks; c += gridDim.x) {
    const int base4 = c << 10;           // 32 rows * 32 float4 per chunk
    float4 v[4];
#pragma unroll
    for (int q = 0; q < 4; ++q) v[q] = X4[base4 + t + 256 * q];

    const int cn = c + gridDim.x;        // prefetch next chunk
    if (cn < nChunks) {
#pragma unroll
      for (int q = 0; q < 4; ++q)
        __builtin_prefetch((const void*)(X4 + (cn << 10) + t + 256 * q), 0, 3);
    }

    __syncthreads();                     // prior compute phase done
#pragma unroll
    for (int q = 0; q < 4; ++q) {
      ((float4*)sX)[(w + 8 * q) * 34 + col4] = v[q];
      csum.x += v[q].x; csum.y += v[q].y; csum.z += v[q].z; csum.w += v[q].w;
    }
    __syncthreads();

#pragma unroll
    for (int kb = 0; kb < 32; kb += 4) {
      v2f a;
      a.x = sX[(kb + 2 * h)     * 136 + 16 * w + l15];
      a.y = sX[(kb + 1 + 2 * h) * 136 + 16 * w + l15];
#pragma unroll
      for (int j = 0; j < 8; ++j) {
        v2f b;
        b.x = sX[(kb + 2 * h)     * 136 + 16 * j + l15];
        b.y = sX[(kb + 1 + 2 * h) * 136 + 16 * j + l15];
        acc[j] = wmma4(a, b, acc[j]);
      }
    }
  }

  // mean partial reduce: LDS first, then one global atomic per column
  atomicAdd(&sm[4 * col4 + 0], csum.x);
  atomicAdd(&sm[4 * col4 + 1], csum.y);
  atomicAdd(&sm[4 * col4 + 2], csum.z);
  atomicAdd(&sm[4 * col4 + 3], csum.w);
  __syncthreads();
  if (t < 128) unsafeAtomicAdd(&mean_acc[t], sm[t]);

  // Gram tiles -> global atomics (C/D layout: reg r, lane -> row r+8h, col l15)
#pragma unroll
  for (int j = 0; j < 8; ++j) {
#pragma unroll
    for (int r = 0; r < 8; ++r) {
      unsafeAtomicAdd(&gram[(16 * w + r + 8 * h) * 128 + 16 * j + l15],
                      acc[j][r]);
    }
  }
}

// =====================================================================
// Kernel 2: mean, Sigma = G/N - mu mu^T + eps I, trace, Sn, P = I
// =====================================================================
__global__ __launch_bounds__(256) void stats_kernel(float* ws, int N)
{
  float* mean_acc = ws + WS_MEAN_ACC;
  float* gram     = ws + WS_GRAM;
  float* mean     = ws + WS_MEAN;
  float* Sn       = ws + WS_SN;
  float* P        = ws + WS_P;

  __shared__ float sm[128];
  __shared__ float red[256];
  __shared__ float srtr;

  const int t = threadIdx.x;
  const float invN = 1.0f / (float)N;

  if (t < 128) {
    float m = mean_acc[t] * invN;
    sm[t] = m;
    mean[t] = m;
  }
  __syncthreads();

  float diag = 0.f;
  if (t < 128) diag = gram[t * 128 + t] * invN - sm[t] * sm[t] + EPS;
  red[t] = diag;
  __syncthreads();
  for (int off = 128; off > 0; off >>= 1) {
    if (t < off) red[t] += red[t + off];
    __syncthreads();
  }
  if (t == 0) {
    float tr = red[0];
    srtr = 1.0f / tr;
    ws[WS_SCAL] = sqrtf(1.0f / tr);
  }
  __syncthreads();
  const float rTr = srtr;

  for (int k = 0; k < 64; ++k) {
    int idx = t + 256 * k;
    int d = idx >> 7, e = idx & 127;
    float sig = gram[idx] * invN - sm[d] * sm[e] + ((d == e) ? EPS : 0.f);
    Sn[idx] = sig * rTr;
    P[idx]  = (d == e) ? 1.f : 0.f;
  }
}

// =====================================================================
// 128x128x128 matmul via f32 WMMA; one 512-thread block, 16 waves x 4 tiles
// storeMode 0: D[row][col] = scale*acc   (row-major)
// storeMode 1: B-fragment order for the output GEMM:
//              element (row=d, col=e) -> D[(e>>2)*512 + d*4 + (e&3)]
// =====================================================================
__device__ inline void mm128(float* __restrict__ D, const float* __restrict__ A,
                             const float* __restrict__ B, int w, int l15, int h,
                             int storeMode, float scale)
{
#pragma unroll
  for (int q = 0; q < 4; ++q) {
    const int tile = w + 16 * q;
    const int i = tile >> 3, j = tile & 7;
    v8f acc = v8zero();
#pragma unroll 8
    for (int kb = 0; kb < 128; kb += 4) {
      v2f a = *(const v2f*)(A + (16 * i + l15) * 128 + kb + 2 * h);
      v2f b;
      b.x = B[(kb + 2 * h)     * 128 + 16 * j + l15];
      b.y = B[(kb + 1 + 2 * h) * 128 + 16 * j + l15];
      acc = wmma4(a, b, acc);
    }
#pragma unroll
    for (int r = 0; r < 8; ++r) {
      const int row = 16 * i + r + 8 * h;
      const int col = 16 * j + l15;
      if (storeMode == 1)
        D[(col >> 2) * 512 + row * 4 + (col & 3)] = scale * acc[r];
      else
        D[row * 128 + col] = scale * acc[r];
    }
  }
}

// Kernel 3: Newton-Schulz P <- 1.5P - 0.5 P^3 Sn (x5),
//           then Wf = fragment-ordered (sqrt(rTr) * rot @ P)
__global__ __launch_bounds__(512) void ns_kernel(float* ws,
                                                 const float* __restrict__ rot)
{
  float* Sn = ws + WS_SN;
  float* P  = ws + WS_P;
  float* T1 = ws + WS_T1;
  float* T2 = ws + WS_T2;
  float* Wf = ws + WS_WF;

  const int t = threadIdx.x, w = t >> 5;
  const int lane = t & 31, l15 = lane & 15, h = lane >> 4;

  for (int it = 0; it < 5; ++it) {
    mm128(T1, P,  P,  w, l15, h, 0, 1.f); gsync();   // T1 = P^2
    mm128(T2, T1, P,  w, l15, h, 0, 1.f); gsync();   // T2 = P^3
    mm128(T1, T2, Sn, w, l15, h, 0, 1.f); gsync();   // T1 = P^3 Sn
    for (int k = t; k < 16384; k += 512)
      P[k] = 1.5f * P[k] - 0.5f * T1[k];
    gsync();
  }
  const float s = ws[WS_SCAL];                 // sqrt(rTr)
  mm128(Wf, rot, P, w, l15, h, 1, s);          // Wf = frag-order(s * rot @ P)
}

// =====================================================================
// Kernel 4: out = (X - mean) @ W^T ; [N,128] x [128,128], K = 128
// block = 256 threads (8 waves); 32-row chunks staged in LDS (pad 132:
// A-fragment column reads hit 32 distinct banks, contiguous K-pairs ->
// ds_load_b64). B fragments are single aligned global b64 loads from the
// fragment-ordered, WGP$-resident Wf.
// =====================================================================
__global__ __launch_bounds__(256) void out_kernel(
    const float* __restrict__ X, const float* __restrict__ ws,
    float* __restrict__ out, int nChunks)
{
  const float* mean = ws + WS_MEAN;
  const float* Wf   = ws + WS_WF;

  __shared__ float sX[32 * 132];

  const int t    = threadIdx.x;
  const int w    = t >> 5;
  const int lane = t & 31, l15 = lane & 15, h = lane >> 4;
  const int col4 = t & 31;
  const int i  = w & 1;            // tile row within chunk (0..1)
  const int j0 = (w >> 1) * 2;     // first of two tile cols (0,2,4,6)

  const float4 m4 = ((const float4*)mean)[col4];
  const float4* X4 = (const float4*)X;

  // per-lane base offsets into fragment-ordered Wf (dwords)
  const int wfBase0 = (16 * j0)       * 4 + l15 * 4 + 2 * h;
  const int wfBase1 = (16 * (j0 + 1)) * 4 + l15 * 4 + 2 * h;

  for (int c = blockIdx.x; c < nChunks; c += gridDim.x) {
    const int base4 = c << 10;
    float4 v[4];
#pragma unroll
    for (int q = 0; q < 4; ++q) v[q] = X4[base4 + t + 256 * q];

    __syncthreads();
#pragma unroll
    for (int q = 0; q < 4; ++q) {
      float4 x = v[q];
      x.x -= m4.x; x.y -= m4.y; x.z -= m4.z; x.w -= m4.w;
      ((float4*)sX)[(w + 8 * q) * 33 + col4] = x;
    }
    __syncthreads();

    v8f acc0 = v8zero(), acc1 = v8zero();
#pragma unroll
    for (int kb = 0; kb < 128; kb += 4) {
      v2f a  = *(const v2f*)&sX[(16 * i + l15) * 132 + kb + 2 * h];
      v2f b0 = *(const v2f*)&Wf[(kb >> 2) * 512 + wfBase0];
      v2f b1 = *(const v2f*)&Wf[(kb >> 2) * 512 + wfBase1];
      acc0 = wmma4(a, b0, acc0);
      acc1 = wmma4(a, b1, acc1);
    }

    const int baseRow = c * 32;
#pragma unroll
    for (int r = 0; r < 8; ++r) {
      const int row = baseRow + 16 * i + r + 8 * h;
      out[row * 128 + 16 * j0 + l15]       = acc0[r];
      out[row * 128 + 16 * (j0 + 1) + l15] = acc1[r];
    }
  }
}

// =====================================================================
extern "C" void kernel_launch(void* const* d_in, const int* in_sizes, int n_in,
                              void* d_out, int out_size, void* d_ws, size_t ws_size,
                              hipStream_t stream)
{
  const float* X   = (const float*)d_in[0];   // [N,128]
  const float* rot = (const float*)d_in[1];   // [1,128,128]
  float* out = (float*)d_out;
  float* ws  = (float*)d_ws;

  const int N = in_sizes[0] / 128;            // 500000
  const int nChunks = N / 32;                 // 15625 (exact)

  hipMemsetAsync(d_ws, 0, (size_t)(128 + 128 * 128) * sizeof(float), stream);

  gram_mean_kernel<<<512, 256, 0, stream>>>(X, ws, nChunks);
  stats_kernel<<<1, 256, 0, stream>>>(ws, N);
  ns_kernel<<<1, 512, 0, stream>>>(ws, rot);
  out_kernel<<<2048, 256, 0, stream>>>(X, ws, out, nChunks);
}